// MaskedTimeAttention_21552145891849
// MI455X (gfx1250) — compile-verified
//
#include <hip/hip_runtime.h>
#include <hip/hip_bf16.h>

typedef __attribute__((ext_vector_type(2))) float v2f;
typedef __attribute__((ext_vector_type(8))) float v8f;

#define B_    4
#define TE    512
#define TD    128
#define DEN   256
#define DDE   512
#define UDIM  128
#define RMF   1000000.0f

// ---------------------------------------------------------------------------
// Prep: en_masked[b,te,:] = en_seq[b,te,:] * (float)mask[b,te]
// float4-vectorized; one pass, keeps the mask multiply out of the GEMM loops.
// ---------------------------------------------------------------------------
__global__ __launch_bounds__(256) void mask_en(
    const float* __restrict__ en_seq, const int* __restrict__ mask,
    float* __restrict__ en_masked)
{
    const int idx = blockIdx.x * blockDim.x + threadIdx.x;  // B_*TE*DEN/4
    const int row = idx / (DEN / 4);                         // b*TE + te
    const float m = (float)mask[row];
    const float4 e = ((const float4*)en_seq)[idx];
    float4 r; r.x = e.x * m; r.y = e.y * m; r.z = e.z * m; r.w = e.w * m;
    ((float4*)en_masked)[idx] = r;
}

// ---------------------------------------------------------------------------
// Batched f32 WMMA GEMM, strip-mined: one wave computes a 16(M) x 64(N) strip
// with 4 accumulators, reusing one A fragment across 4 independent
// V_WMMA_F32_16X16X4_F32 ops per K-step (pipelines in the XDL).
// A-layout (ISA 7.12.2, 32-bit A 16x4): lane%16 = M, lane/16 picks K{0,1}/{2,3}
// B symmetric: lane%16 = N.  C/D: vgpr r, lanegroup g -> M = r + 8g.
// All offsets 32-bit (matrices < 4M elements) -> saddr + voffset addressing.
// ---------------------------------------------------------------------------
__global__ __launch_bounds__(128) void wmma_gemm_f32(
    const float* __restrict__ A, const float* __restrict__ Bm, float* __restrict__ C,
    int M, int N, int K, int lda, int ldb, int ldc,
    int sAb, int sBb, int sCb)
{
    const float* Ab = A  + blockIdx.y * sAb;
    const float* Bb = Bm + blockIdx.y * sBb;
    float*       Cb = C  + blockIdx.y * sCb;

    const int stripsN = N >> 6;                       // strips of 64 columns
    const int total   = (M >> 4) * stripsN;
    const int t = blockIdx.x * (blockDim.x >> 5) + (threadIdx.x >> 5);
    if (t >= total) return;
    const int tm = t / stripsN;
    const int ts = t - tm * stripsN;

    const int lane = threadIdx.x & 31;
    const int l = lane & 15;                          // M for A, N for B/C
    const int g = lane >> 4;                          // K-half selector
    const int m = tm * 16 + l;
    const int n = ts * 64 + l;

    const float* arow = Ab + m * lda;
    v8f acc0 = {}, acc1 = {}, acc2 = {}, acc3 = {};

    #pragma unroll 4
    for (int k0 = 0; k0 < K; k0 += 4) {
        const int ka = k0 + 2 * g;
        const v2f a = *(const v2f*)(arow + ka);       // 8B-aligned (ka even)
        const float* bp0 = Bb + ka * ldb + n;
        const float* bp1 = bp0 + ldb;
        v2f b0, b1, b2, b3;
        b0.x = bp0[0];  b0.y = bp1[0];
        b1.x = bp0[16]; b1.y = bp1[16];
        b2.x = bp0[32]; b2.y = bp1[32];
        b3.x = bp0[48]; b3.y = bp1[48];
        // (neg_a, A, neg_b, B, c_mod, C, reuse_a, reuse_b)
        acc0 = __builtin_amdgcn_wmma_f32_16x16x4_f32(false, a, false, b0, (short)0, acc0, false, false);
        acc1 = __builtin_amdgcn_wmma_f32_16x16x4_f32(false, a, false, b1, (short)0, acc1, false, false);
        acc2 = __builtin_amdgcn_wmma_f32_16x16x4_f32(false, a, false, b2, (short)0, acc2, false, false);
        acc3 = __builtin_amdgcn_wmma_f32_16x16x4_f32(false, a, false, b3, (short)0, acc3, false, false);
    }

    float* crow = Cb + (tm * 16 + 8 * g) * ldc + n;
    #pragma unroll
    for (int r = 0; r < 8; ++r) {
        crow[r * ldc +  0] = acc0[r];
        crow[r * ldc + 16] = acc1[r];
        crow[r * ldc + 32] = acc2[r];
        crow[r * ldc + 48] = acc3[r];
    }
}

// ---------------------------------------------------------------------------
// Fused scores + softmax: one block per (b, td).
// mu[te] = sum_u tanh(en_p[b,te,u] + de_p[b,td,u]) * nu[u]  + (mask-1)*1e6
// alphas[b,td,:] = softmax_te(mu).  The [B,Td,Te,U] tensor never materialized.
// ---------------------------------------------------------------------------
__global__ __launch_bounds__(256) void scores_softmax(
    const float* __restrict__ en_p, const float* __restrict__ de_p,
    const float* __restrict__ nu, const int* __restrict__ mask,
    float* __restrict__ alphas)
{
    const int b  = blockIdx.x >> 7;        // TD = 128
    const int td = blockIdx.x & (TD - 1);

    __shared__ float s_de[UDIM];
    __shared__ float s_nu[UDIM];
    __shared__ float red[256];

    const int tid = threadIdx.x;
    if (tid < UDIM) {
        s_de[tid] = de_p[(b * TD + td) * UDIM + tid];
        s_nu[tid] = nu[tid];
    }
    __syncthreads();

    float mu0 = 0.f, mu1 = 0.f;
    #pragma unroll
    for (int half = 0; half < 2; ++half) {
        const int te = tid + half * 256;
        const float4* ep = (const float4*)(en_p + (b * TE + te) * UDIM);
        float s = 0.f;
        #pragma unroll 4
        for (int u = 0; u < UDIM / 4; ++u) {
            const float4 e = ep[u];
            s += tanhf(e.x + s_de[4 * u + 0]) * s_nu[4 * u + 0];
            s += tanhf(e.y + s_de[4 * u + 1]) * s_nu[4 * u + 1];
            s += tanhf(e.z + s_de[4 * u + 2]) * s_nu[4 * u + 2];
            s += tanhf(e.w + s_de[4 * u + 3]) * s_nu[4 * u + 3];
        }
        s += ((float)mask[b * TE + te] - 1.0f) * RMF;
        if (half == 0) mu0 = s; else mu1 = s;
    }

    red[tid] = fmaxf(mu0, mu1);
    __syncthreads();
    for (int off = 128; off > 0; off >>= 1) {
        if (tid < off) red[tid] = fmaxf(red[tid], red[tid + off]);
        __syncthreads();
    }
    const float mx = red[0];
    __syncthreads();

    const float e0 = __expf(mu0 - mx);
    const float e1 = __expf(mu1 - mx);
    red[tid] = e0 + e1;
    __syncthreads();
    for (int off = 128; off > 0; off >>= 1) {
        if (tid < off) red[tid] += red[tid + off];
        __syncthreads();
    }
    const float inv = 1.0f / red[0];

    float* arow = alphas + (b * TD + td) * TE;
    arow[tid]       = e0 * inv;
    arow[tid + 256] = e1 * inv;
}

// ---------------------------------------------------------------------------
// Copy de_seq into out[:, :, 0:512] (float4-vectorized).
// ---------------------------------------------------------------------------
__global__ __launch_bounds__(256) void copy_de(
    const float* __restrict__ de_seq, float* __restrict__ out)
{
    const int idx = blockIdx.x * blockDim.x + threadIdx.x;  // B_*TD*DDE/4 float4s
    const int row = idx / (DDE / 4);
    const int f4  = idx - row * (DDE / 4);
    float4* dst = (float4*)(out + row * (DDE + DEN));
    dst[f4] = ((const float4*)de_seq)[idx];
}

extern "C" void kernel_launch(void* const* d_in, const int* in_sizes, int n_in,
                              void* d_out, int out_size, void* d_ws, size_t ws_size,
                              hipStream_t stream)
{
    const float* en_seq = (const float*)d_in[0];   // [4,512,256]
    const float* de_seq = (const float*)d_in[1];   // [4,128,512]
    const int*   mask   = (const int*)  d_in[2];   // [4,512]
    const float* w_en   = (const float*)d_in[3];   // [256,128]
    const float* w_de   = (const float*)d_in[4];   // [512,128]
    const float* nu     = (const float*)d_in[5];   // [128,1]
    float* out = (float*)d_out;                    // [4,128,768]

    float* ws        = (float*)d_ws;               // ~4.3 MB used, L2-resident
    float* en_masked = ws;                                   // B_*TE*DEN = 524288 f
    float* en_p      = en_masked + B_ * TE * DEN;            // B_*TE*U   = 262144 f
    float* de_p      = en_p      + B_ * TE * UDIM;           // B_*TD*U   =  65536 f
    float* alphas    = de_p      + B_ * TD * UDIM;           // B_*TD*TE  = 262144 f

    // 0) en_masked = mask ⊙ en_seq  (used as GEMM-1 A and GEMM-4 B)
    mask_en<<<(B_ * TE * DEN / 4) / 256, 256, 0, stream>>>(en_seq, mask, en_masked);

    // 1) en_p = en_masked @ w_en    [2048x256]·[256x128]
    {
        const int M = B_ * TE, N = UDIM, K = DEN;
        const int waves = (M / 16) * (N / 64);     // 128*2 = 256
        wmma_gemm_f32<<<dim3((waves + 3) / 4, 1), 128, 0, stream>>>(
            en_masked, w_en, en_p, M, N, K, K, N, N, 0, 0, 0);
    }
    // 2) de_p = de_seq @ w_de       [512x512]·[512x128]
    {
        const int M = B_ * TD, N = UDIM, K = DDE;
        const int waves = (M / 16) * (N / 64);     // 32*2 = 64
        wmma_gemm_f32<<<dim3((waves + 3) / 4, 1), 128, 0, stream>>>(
            de_seq, w_de, de_p, M, N, K, K, N, N, 0, 0, 0);
    }
    // 3) fused tanh-scores + mask bias + softmax -> alphas
    scores_softmax<<<B_ * TD, 256, 0, stream>>>(en_p, de_p, nu, mask, alphas);

    // 4) out[:,:,512:768] = alphas @ en_masked, batched over B
    {
        const int M = TD, N = DEN, K = TE;
        const int waves = (M / 16) * (N / 64);     // 8*4 = 32
        wmma_gemm_f32<<<dim3((waves + 3) / 4, B_), 128, 0, stream>>>(
            alphas, en_masked, out + DDE, M, N, K, TE, DEN, DDE + DEN,
            TD * TE, TE * DEN, TD * (DDE + DEN));
    }
    // 5) out[:,:,0:512] = de_seq
    copy_de<<<(B_ * TD * DDE / 4) / 256, 256, 0, stream>>>(de_seq, out);
}